// GCN__43791486550061
// MI455X (gfx1250) — compile-verified
//
#include <hip/hip_runtime.h>
#include <hip/hip_bf16.h>

typedef __attribute__((ext_vector_type(2))) float v2f;
typedef __attribute__((ext_vector_type(8))) float v8f;

// ---------------------------------------------------------------------------
// Small elementwise helpers
// ---------------------------------------------------------------------------
__global__ void k_fill(float* __restrict__ p, float v, int n) {
    int i = blockIdx.x * blockDim.x + threadIdx.x;
    if (i < n) p[i] = v;
}

__global__ void k_deg_edges(const long long* __restrict__ dst, float* __restrict__ deg, int E) {
    int e = blockIdx.x * blockDim.x + threadIdx.x;
    if (e < E) atomicAdd(&deg[(int)dst[e]], 1.0f);
}

__global__ void k_rsqrt(float* __restrict__ d, int n) {
    int i = blockIdx.x * blockDim.x + threadIdx.x;
    if (i < n) {
        float v = d[i];
        d[i] = (v > 0.0f) ? rsqrtf(v) : 0.0f;
    }
}

__global__ void k_relu(float* __restrict__ p, int n) {
    int i = blockIdx.x * blockDim.x + threadIdx.x;
    if (i < n) p[i] = fmaxf(p[i], 0.0f);
}

// ---------------------------------------------------------------------------
// WMMA f32 GEMM: C[M,N] = A[M,K] @ B[K,N], row-major, one wave per 16x16 tile.
// Uses V_WMMA_F32_16X16X4_F32 (CDNA5, wave32).
// A frag (16x4):  lanes 0-15 hold K=k0,k0+1 ; lanes 16-31 hold K=k0+2,k0+3
// B frag (4x16):  same split, striped across lanes by column
// C/D (16x16):    VGPR r: lanes 0-15 -> M=r, lanes 16-31 -> M=r+8
// ---------------------------------------------------------------------------
__global__ void k_gemm_wmma(const float* __restrict__ A, const float* __restrict__ B,
                            float* __restrict__ C, int Mtiles, int Ntiles, int K, int N) {
    int wave = blockIdx.x * (blockDim.x >> 5) + (threadIdx.x >> 5);
    if (wave >= Mtiles * Ntiles) return;   // uniform per-wave: EXEC stays all-1s
    int tm = wave / Ntiles;
    int tn = wave % Ntiles;

    int lane = threadIdx.x & 31;
    int half = lane >> 4;     // 0: K pair {0,1}, 1: K pair {2,3}
    int sub  = lane & 15;

    const float* arow = A + (long long)(tm * 16 + sub) * K;
    const float* bcol = B + tn * 16 + sub;

    v8f acc = {0.f, 0.f, 0.f, 0.f, 0.f, 0.f, 0.f, 0.f};
    for (int k0 = 0; k0 < K; k0 += 4) {
        int ka = k0 + (half << 1);
        v2f a;
        a.x = arow[ka];
        a.y = arow[ka + 1];
        v2f b;
        b.x = bcol[(long long)ka * N];
        b.y = bcol[(long long)(ka + 1) * N];
        acc = __builtin_amdgcn_wmma_f32_16x16x4_f32(
            /*neg_a=*/false, a, /*neg_b=*/false, b,
            /*c_mod=*/(short)0, acc, /*reuse_a=*/false, /*reuse_b=*/false);
    }

    float* crow = C + (long long)(tm * 16 + (half << 3)) * N + tn * 16 + sub;
#pragma unroll
    for (int r = 0; r < 8; r++) crow[(long long)r * N] = acc[r];
}

// ---------------------------------------------------------------------------
// Aggregation: agg[i,f] = hlin[i,f]*dinv[i]^2 + bias[f]   (self-loop + bias)
// then edge atomics: agg[d,f] += hlin[s,f] * dinv[s]*dinv[d]
// ---------------------------------------------------------------------------
template <int F>
__global__ void k_agg_init(const float* __restrict__ hlin, const float* __restrict__ dinv,
                           const float* __restrict__ bias, float* __restrict__ agg, int n) {
    int idx = blockIdx.x * blockDim.x + threadIdx.x;
    if (idx >= n * F) return;
    int i = idx / F;
    int f = idx % F;
    float di = dinv[i];
    agg[idx] = hlin[idx] * di * di + bias[f];
}

template <int F, int FPT>
__global__ void k_agg_edges(const long long* __restrict__ src, const long long* __restrict__ dst,
                            const float* __restrict__ hlin, const float* __restrict__ dinv,
                            float* __restrict__ agg, int E) {
    constexpr int TPE = F / FPT;  // threads per edge (power of two)
    long long idx = (long long)blockIdx.x * blockDim.x + threadIdx.x;
    int e = (int)(idx / TPE);
    int t = (int)(idx % TPE);
    if (e >= E) return;
    int s = (int)src[e];
    int d = (int)dst[e];
    float w = dinv[s] * dinv[d];
    int fb = t * FPT;
    const float* hp = hlin + (long long)s * F + fb;
    float* ap = agg + (long long)d * F + fb;
#pragma unroll
    for (int j = 0; j < FPT; j++) atomicAdd(&ap[j], hp[j] * w);
}

// ---------------------------------------------------------------------------
// In-place row-wise log_softmax over 16 classes (one thread per node)
// ---------------------------------------------------------------------------
__global__ void k_logsoftmax16(float* __restrict__ out, int n) {
    int i = blockIdx.x * blockDim.x + threadIdx.x;
    if (i >= n) return;
    float* row = out + (long long)i * 16;
    float v[16];
    float m = -3.402823466e38f;
#pragma unroll
    for (int f = 0; f < 16; f++) {
        v[f] = row[f];
        m = fmaxf(m, v[f]);
    }
    float s = 0.0f;
#pragma unroll
    for (int f = 0; f < 16; f++) s += __expf(v[f] - m);
    float lse = m + __logf(s);
#pragma unroll
    for (int f = 0; f < 16; f++) row[f] = v[f] - lse;
}

// ---------------------------------------------------------------------------
// Launch
// ---------------------------------------------------------------------------
extern "C" void kernel_launch(void* const* d_in, const int* in_sizes, int n_in,
                              void* d_out, int out_size, void* d_ws, size_t ws_size,
                              hipStream_t stream) {
    const float* x  = (const float*)d_in[0];         // [N,256]
    const long long* ei = (const long long*)d_in[1]; // [2,E] int64
    const float* W1 = (const float*)d_in[2];         // [256,128]
    const float* b1 = (const float*)d_in[3];         // [128]
    const float* W2 = (const float*)d_in[4];         // [128,16]
    const float* b2 = (const float*)d_in[5];         // [16]
    float* out = (float*)d_out;                      // [N,16]

    const int N = in_sizes[0] / 256;   // 50000
    const int E = in_sizes[1] / 2;     // 800000
    const long long* src = ei;
    const long long* dst = ei + E;

    // Workspace layout (256B aligned blocks)
    char* ws = (char*)d_ws;
    size_t off = 0;
    auto alloc = [&](size_t bytes) {
        char* p = ws + off;
        off += (bytes + 255) & ~(size_t)255;
        return p;
    };
    float* dinv = (float*)alloc((size_t)N * 4);            // degree then 1/sqrt(deg)
    float* hlin = (float*)alloc((size_t)N * 128 * 4);      // x @ W1
    float* agg1 = (float*)alloc((size_t)N * 128 * 4);      // layer-1 aggregate / h
    float* olin = (float*)alloc((size_t)N * 16 * 4);       // h @ W2
    (void)ws_size;

    const int TB = 256;
    auto blk = [](long long work, int tb) { return (unsigned)((work + tb - 1) / tb); };

    // 1) symmetric normalization: deg (with self-loop) -> rsqrt
    k_fill<<<blk(N, TB), TB, 0, stream>>>(dinv, 1.0f, N);
    k_deg_edges<<<blk(E, TB), TB, 0, stream>>>(dst, dinv, E);
    k_rsqrt<<<blk(N, TB), TB, 0, stream>>>(dinv, N);

    // 2) h_lin = x @ W1   (3125 x 8 tiles of 16x16, K=256)
    {
        int mt = N / 16, nt = 128 / 16;
        long long waves = (long long)mt * nt;
        k_gemm_wmma<<<blk(waves * 32, 128), 128, 0, stream>>>(x, W1, hlin, mt, nt, 256, 128);
    }

    // 3) agg1 = b1 + self-loop + edge scatter; then ReLU
    k_agg_init<128><<<blk((long long)N * 128, TB), TB, 0, stream>>>(hlin, dinv, b1, agg1, N);
    k_agg_edges<128, 4><<<blk((long long)E * 32, TB), TB, 0, stream>>>(src, dst, hlin, dinv, agg1, E);
    k_relu<<<blk((long long)N * 128, TB), TB, 0, stream>>>(agg1, N * 128);

    // 4) o_lin = h @ W2   (3125 x 1 tiles, K=128)
    {
        int mt = N / 16, nt = 1;
        long long waves = (long long)mt * nt;
        k_gemm_wmma<<<blk(waves * 32, 128), 128, 0, stream>>>(agg1, W2, olin, mt, nt, 128, 16);
    }

    // 5) second aggregation straight into d_out (bias folded into init)
    k_agg_init<16><<<blk((long long)N * 16, TB), TB, 0, stream>>>(olin, dinv, b2, out, N);
    k_agg_edges<16, 1><<<blk((long long)E * 16, TB), TB, 0, stream>>>(src, dst, olin, dinv, out, E);

    // 6) log_softmax per row, in place
    k_logsoftmax16<<<blk(N, 128), 128, 0, stream>>>(out, N);
}